// RGCN_43765716746409
// MI455X (gfx1250) — compile-verified
//
#include <hip/hip_runtime.h>

#define NN 50000
#define DD 128
#define RR 8
#define BB 8
#define EE 625000
#define LL 2

typedef __attribute__((ext_vector_type(2))) float v2f;
typedef __attribute__((ext_vector_type(4))) float v4f;
typedef __attribute__((ext_vector_type(8))) float v8f;

// ---- CDNA5 async global->LDS helpers (ASYNCcnt-tracked) --------------------
__device__ __forceinline__ unsigned lds_off(const void* p) {
  // generic pointers to LDS carry the wave-relative LDS offset in the low 32b
  return (unsigned)(size_t)p;
}
__device__ __forceinline__ void async_g2l_b128(unsigned ldsaddr, const void* g) {
  asm volatile("global_load_async_to_lds_b128 %0, %1, off"
               :: "v"(ldsaddr), "v"(g) : "memory");
}
__device__ __forceinline__ void wait_async0() {
  asm volatile("s_wait_asynccnt 0" ::: "memory");
}

// ---------------- kernel 1: W[r] = sum_b comp[r,b] * bases[b] ----------------
__global__ void k_comp_w(const float* __restrict__ comp,
                         const float* __restrict__ bases,
                         float* __restrict__ Wout) {
  int t = blockIdx.x * blockDim.x + threadIdx.x;  // over R*D*D = 131072
  if (t >= RR * DD * DD) return;
  int r  = t >> 14;       // / (128*128)
  int dd = t & 16383;
  float acc = 0.f;
#pragma unroll
  for (int b = 0; b < BB; ++b)
    acc += comp[r * BB + b] * bases[(size_t)b * DD * DD + dd];
  Wout[t] = acc;
}

// ---------------- kernel 2: 16-row strip GEMM via V_WMMA_F32_16X16X4_F32 ----
// C_strip(16x128) = H_strip(16x128) @ W[blockIdx.y](128x128) (+bias)
// out element (row, col) -> out[row*out_row_stride + blockIdx.y*out_rel_off + col]
#define HPAD 132   // 16 rows * 4-bank skew -> 16 distinct banks for A frags
#define WPAD 144   // half-wave row skew of 32 banks -> no B-frag conflicts
#define KCH  32    // K chunk staged in LDS per iteration (double-buffered)

template <bool NT>
__global__ __launch_bounds__(128) void k_gemm16(
    const float* __restrict__ H, const float* __restrict__ Wall,
    const float* __restrict__ bias, float* __restrict__ out,
    int out_row_stride, int out_rel_off) {
  __shared__ float hT[16 * HPAD];
  __shared__ float wT[2][KCH * WPAD];

  const int tid    = threadIdx.x;
  const int lane   = tid & 31;
  const int wave   = tid >> 5;           // 0..3
  const int strip  = blockIdx.x;         // node strip (16 rows)
  const int rel    = blockIdx.y;
  const float* __restrict__ Wm = Wall + (size_t)rel * DD * DD;

  const int rowSel = lane & 15;
  const int koff   = (lane >> 4) * 2;    // lanes 0-15 -> K {0,1}; 16-31 -> {2,3}
  const int n0     = wave * 32;          // this wave covers cols [n0, n0+32)

  // async-stage h strip (16x128, 512 float4 -> 4/thread) and W chunk 0
  {
    const float4* hsrc = (const float4*)(H + (size_t)strip * 16 * DD);
#pragma unroll
    for (int i = 0; i < 4; ++i) {
      int e = tid + i * 128, rr = e >> 5, cc = (e & 31) * 4;
      async_g2l_b128(lds_off(&hT[rr * HPAD + cc]), hsrc + e);
    }
    const float4* wsrc = (const float4*)Wm;
#pragma unroll
    for (int i = 0; i < 8; ++i) {
      int e = tid + i * 128, rr = e >> 5, cc = (e & 31) * 4;
      async_g2l_b128(lds_off(&wT[0][rr * WPAD + cc]), wsrc + e);
    }
  }
  wait_async0();
  __syncthreads();

  v8f acc0 = {};
  v8f acc1 = {};

#pragma unroll
  for (int kc = 0; kc < DD / KCH; ++kc) {
    const int cur = kc & 1;

    // prefetch next W chunk into the other LDS buffer while we compute
    if (kc + 1 < DD / KCH) {
      const float4* wn = (const float4*)(Wm + (size_t)(kc + 1) * KCH * DD);
#pragma unroll
      for (int i = 0; i < 8; ++i) {
        int e = tid + i * 128, rr = e >> 5, cc = (e & 31) * 4;
        async_g2l_b128(lds_off(&wT[1 - cur][rr * WPAD + cc]), wn + e);
      }
    }

    const float* __restrict__ wTc = wT[cur];
#pragma unroll
    for (int kk = 0; kk < KCH / 4; ++kk) {
      // A fragment: 16x4, lane holds row=rowSel, K = 4*kk + koff + {0,1}
      v2f a = *(const v2f*)&hT[rowSel * HPAD + kc * KCH + kk * 4 + koff];
      // B fragments: 4x16, rows striped across lanes
      int kb = kk * 4 + koff;
      v2f b0, b1;
      b0.x = wTc[(kb + 0) * WPAD + n0 + rowSel];
      b0.y = wTc[(kb + 1) * WPAD + n0 + rowSel];
      b1.x = wTc[(kb + 0) * WPAD + n0 + 16 + rowSel];
      b1.y = wTc[(kb + 1) * WPAD + n0 + 16 + rowSel];
      acc0 = __builtin_amdgcn_wmma_f32_16x16x4_f32(false, a, false, b0,
                                                   (short)0, acc0, false, false);
      acc1 = __builtin_amdgcn_wmma_f32_16x16x4_f32(false, a, false, b1,
                                                   (short)0, acc1, false, false);
    }
    wait_async0();
    __syncthreads();
  }

  // store: C/D layout — VGPR i: lanes0-15 -> M=i, lanes16-31 -> M=i+8
  const int col0 = n0 + rowSel;
  const int col1 = n0 + 16 + rowSel;
  float bv0 = 0.f, bv1 = 0.f;
  if (bias) { bv0 = bias[col0]; bv1 = bias[col1]; }
  const int mhi = (lane >> 4) * 8;
#pragma unroll
  for (int i = 0; i < 8; ++i) {
    size_t row = (size_t)strip * 16 + i + mhi;
    float* o = out + row * (size_t)out_row_stride + (size_t)rel * out_rel_off;
    if (NT) {
      // xW is written once / read once and exceeds L2 -> keep it non-temporal
      __builtin_nontemporal_store(acc0[i] + bv0, &o[col0]);
      __builtin_nontemporal_store(acc1[i] + bv1, &o[col1]);
    } else {
      o[col0] = acc0[i] + bv0;
      o[col1] = acc1[i] + bv1;
    }
  }
}

// ---------------- kernel 3: edge gather + atomic scatter-sum ----------------
// out[dst[e]] += xW[src[e], etype[e], :]   (32 lanes x float4 per edge)
__global__ void k_edge_scatter(const float* __restrict__ xW,
                               const int* __restrict__ src,
                               const int* __restrict__ dst,
                               const int* __restrict__ et,
                               float* __restrict__ out) {
  long long t = (long long)blockIdx.x * blockDim.x + threadIdx.x;
  int e    = (int)(t >> 5);
  int lane = (int)(t & 31);
  if (e >= EE) return;
  int s = src[e], d = dst[e], r = et[e];
  const v4f* m = (const v4f*)(xW + ((size_t)s * RR + r) * DD);
  // single-use 320MB/layer stream: non-temporal so L2 stays hot for `out`
  v4f v = __builtin_nontemporal_load(m + lane);
  float* o = out + (size_t)d * DD + lane * 4;
  atomicAdd(o + 0, v.x);
  atomicAdd(o + 1, v.y);
  atomicAdd(o + 2, v.z);
  atomicAdd(o + 3, v.w);
}

// ---------------- host launcher ----------------
extern "C" void kernel_launch(void* const* d_in, const int* in_sizes, int n_in,
                              void* d_out, int out_size, void* d_ws, size_t ws_size,
                              hipStream_t stream) {
  const float* x      = (const float*)d_in[0];
  const int*   src    = (const int*)d_in[1];
  const int*   dst    = (const int*)d_in[2];
  const int*   etype  = (const int*)d_in[3];
  const float* bases  = (const float*)d_in[4];  // [L,B,D,D]
  const float* comp   = (const float*)d_in[5];  // [L,R,B]
  const float* loop_w = (const float*)d_in[6];  // [L,D,D]
  const float* bias   = (const float*)d_in[7];  // [L,D]

  float* ws   = (float*)d_ws;
  float* xW   = ws;                                   // N*R*D floats (204.8 MB)
  float* Wbuf = xW + (size_t)NN * RR * DD;            // R*D*D floats
  float* h1   = Wbuf + (size_t)RR * DD * DD;          // N*D floats

  const int strips = NN / 16;  // 3125

  for (int l = 0; l < LL; ++l) {
    const float* hin  = (l == 0) ? x : h1;
    float*       hout = (l == LL - 1) ? (float*)d_out : h1;

    // W[r] = sum_b comp[r,b] * bases[b]
    k_comp_w<<<(RR * DD * DD + 255) / 256, 256, 0, stream>>>(
        comp + (size_t)l * RR * BB, bases + (size_t)l * BB * DD * DD, Wbuf);

    // hout = hin @ loop_w + bias   (initializes output; scatter adds on top)
    k_gemm16<false><<<dim3(strips, 1), 128, 0, stream>>>(
        hin, loop_w + (size_t)l * DD * DD, bias + (size_t)l * DD,
        hout, DD, 0);

    // xW[n,r,:] = hin @ W[r]   (non-temporal stores: xW > L2)
    k_gemm16<true><<<dim3(strips, RR), 128, 0, stream>>>(
        hin, Wbuf, nullptr, xW, RR * DD, DD);

    // hout[dst] += xW[src, etype]
    long long threads = (long long)EE * 32;
    k_edge_scatter<<<(unsigned)((threads + 255) / 256), 256, 0, stream>>>(
        xW, src, dst, etype, hout);
  }
}